// Surv_Loss_17806934409561
// MI455X (gfx1250) — compile-verified
//
#include <hip/hip_runtime.h>
#include <hip/hip_bf16.h>

// CDNA5 / gfx1250, wave32.
typedef __attribute__((ext_vector_type(2))) float v2f;   // WMMA 16x16x4 A/B operand (64 f32 / 32 lanes)
typedef __attribute__((ext_vector_type(4))) float f32x4; // b128 vector load
typedef __attribute__((ext_vector_type(8))) float v8f;   // WMMA 16x16 f32 C/D (256 f32 / 32 lanes)

#define WAVES_PER_BLOCK 32   // 1024 threads; one row per wave32

// ---------------------------------------------------------------------------
// Kernel 1: one wave32 per row. Reads ONLY the bytes the math needs:
//   status==1 -> single dword  (y_pred[row, v])
//   status==0 -> prefix [0, v) via coalesced non-temporal b128 loads
// Deterministic fixed-order reduction: lane accum -> shfl_xor tree -> LDS tree.
// One fp32 partial per block into d_ws.
// ---------------------------------------------------------------------------
__global__ __launch_bounds__(WAVES_PER_BLOCK * 32)
void surv_loss_partials(const float* __restrict__ y_pred,
                        const int*   __restrict__ y,
                        const int*   __restrict__ status,
                        float*       __restrict__ partial,
                        int B, int T)
{
    const int lane = threadIdx.x & 31;
    const int wave = threadIdx.x >> 5;               // 0..31
    const int row  = blockIdx.x * WAVES_PER_BLOCK + wave;

    float loss = 0.0f;
    if (row < B) {
        const int v  = y[row];
        const int st = status[row];
        const float* __restrict__ rp = y_pred + (size_t)row * (size_t)T;

        if (st == 1) {
            // -log(y_pred[row, v]) : 4 bytes only.
            if (lane == 0) loss = -__builtin_logf(rp[v]);
        } else {
            // exclusive prefix sum of rp[0 .. v-1].
            // base = lane*4 + k*128 is a multiple of 4; base < v <= T-1 ensures
            // base+3 <= T-1, so the full b128 load stays inside the row.
            float s = 0.0f;
            for (int base = lane * 4; base < v; base += 128) {
                const f32x4 d =
                    __builtin_nontemporal_load((const f32x4*)(rp + base)); // global_load_b128 nt
                s += (base + 0 < v) ? d.x : 0.0f;
                s += (base + 1 < v) ? d.y : 0.0f;
                s += (base + 2 < v) ? d.z : 0.0f;
                s += (base + 3 < v) ? d.w : 0.0f;
            }
            // wave32 butterfly reduction (fixed order -> deterministic)
            #pragma unroll
            for (int off = 16; off > 0; off >>= 1)
                s += __shfl_xor(s, off, 32);
            if (lane == 0) loss = -__builtin_logf(1.0f - s);
        }
    }

    __shared__ float sh[WAVES_PER_BLOCK];
    if (lane == 0) sh[wave] = loss;
    __syncthreads();

    if (threadIdx.x < 32) {
        float t = sh[threadIdx.x];
        #pragma unroll
        for (int off = 16; off > 0; off >>= 1)
            t += __shfl_xor(t, off, 32);
        if (threadIdx.x == 0) partial[blockIdx.x] = t;
    }
}

// ---------------------------------------------------------------------------
// Kernel 2: single wave32 reduces N block-partials with V_WMMA_F32_16X16X4_F32.
// A = 16x4 chunk of partials, B = ones(4x16)  =>  D[m,n] = C[m,n] + sum_k A[m,k].
// Four independent accumulator chains hide XDL latency; all columns of D are
// identical, so we fold column 0 (lanes 0 and 16). Fully fp32, fixed order.
// ---------------------------------------------------------------------------
__global__ __launch_bounds__(32)
void surv_loss_wmma_reduce(const float* __restrict__ partial,
                           float*       __restrict__ out,
                           int N)
{
    const int lane = threadIdx.x;          // 0..31, EXEC all ones (WMMA requirement)
    const int m    = lane & 15;            // A-matrix row for this lane
    const int koff = (lane >> 4) * 2;      // lanes 0-15: K=0,1 ; lanes 16-31: K=2,3

    const v2f bOnes = {1.0f, 1.0f};        // B = ones(4x16)
    v8f c0 = {}, c1 = {}, c2 = {}, c3 = {};

    const int body = (N / 256) * 256;      // 4 chains * 64 elements per step
    for (int base = 0; base < body; base += 256) {
        const float* p0 = partial + base +   0 + m * 4 + koff;
        const float* p1 = partial + base +  64 + m * 4 + koff;
        const float* p2 = partial + base + 128 + m * 4 + koff;
        const float* p3 = partial + base + 192 + m * 4 + koff;
        v2f a0 = {p0[0], p0[1]};
        v2f a1 = {p1[0], p1[1]};
        v2f a2 = {p2[0], p2[1]};
        v2f a3 = {p3[0], p3[1]};
        c0 = __builtin_amdgcn_wmma_f32_16x16x4_f32(false, a0, false, bOnes, (short)0, c0, false, false);
        c1 = __builtin_amdgcn_wmma_f32_16x16x4_f32(false, a1, false, bOnes, (short)0, c1, false, false);
        c2 = __builtin_amdgcn_wmma_f32_16x16x4_f32(false, a2, false, bOnes, (short)0, c2, false, false);
        c3 = __builtin_amdgcn_wmma_f32_16x16x4_f32(false, a3, false, bOnes, (short)0, c3, false, false);
    }

    // Combine chains, then fold this lane's 8 D rows (column N = lane%16).
    v8f c = c0 + c1 + c2 + c3;
    float s = c[0] + c[1] + c[2] + c[3] + c[4] + c[5] + c[6] + c[7];
    // lane 0 holds D[0..7, 0]; lane 16 holds D[8..15, 0].
    const float sHi = __shfl(s, 16, 32);

    if (lane == 0) {
        float tail = 0.0f;
        for (int i = body; i < N; ++i) tail += partial[i];   // deterministic tail
        out[0] = s + sHi + tail;
    }
}

// ---------------------------------------------------------------------------
extern "C" void kernel_launch(void* const* d_in, const int* in_sizes, int n_in,
                              void* d_out, int out_size, void* d_ws, size_t ws_size,
                              hipStream_t stream) {
    const float* y_pred = (const float*)d_in[0];
    const int*   y      = (const int*)d_in[1];
    const int*   status = (const int*)d_in[2];
    float*       out    = (float*)d_out;

    const int B = in_sizes[1];              // 262144
    const int T = in_sizes[0] / B;          // 512

    float* partial = (float*)d_ws;          // (B+31)/32 floats = 32 KB for B=262144
    const int nBlocks = (B + WAVES_PER_BLOCK - 1) / WAVES_PER_BLOCK;

    surv_loss_partials<<<nBlocks, WAVES_PER_BLOCK * 32, 0, stream>>>(
        y_pred, y, status, partial, B, T);
    surv_loss_wmma_reduce<<<1, 32, 0, stream>>>(partial, out, nBlocks);
}